// RnntLoss_32865089749232
// MI455X (gfx1250) — compile-verified
//
#include <hip/hip_runtime.h>
#include <cstdint>
#include <cstddef>

// Problem constants (match reference)
constexpr int B = 8, T = 200, U = 50, V = 1024;
constexpr int NB = T * U;                 // 10000 blank entries per batch
constexpr int NE = T * (U - 1);           // 9800 emit entries per batch
constexpr int EM_STRIDE = (NE + 3) & ~3;  // 9800: keep per-batch emit base 16B-aligned

#define DEV_INLINE static __device__ __forceinline__

// ---- CDNA5 async global->LDS copy (guarded; falls back to plain LDS stores) ----
#if defined(__has_builtin)
#  if __has_builtin(__builtin_amdgcn_global_load_async_to_lds_b128)
#    define HAVE_ASYNC_B128 1
#  endif
#  if __has_builtin(__builtin_amdgcn_global_load_async_to_lds_b32)
#    define HAVE_ASYNC_B32 1
#  endif
#  if __has_builtin(__builtin_amdgcn_s_wait_asynccnt)
#    define HAVE_WAIT_ASYNC 1
#  endif
#endif

// Typed, address-space-qualified pointers matching the builtin signatures:
//   b128: (v4i addrspace(1)*, v4i addrspace(3)*, imm, imm)
//   b32 : (int addrspace(1)*, int addrspace(3)*, imm, imm)
typedef int v4i __attribute__((ext_vector_type(4)));
typedef __attribute__((address_space(1))) v4i* gptr_v4i;
typedef __attribute__((address_space(3))) v4i* lptr_v4i;
typedef __attribute__((address_space(1))) int* gptr_i32;
typedef __attribute__((address_space(3))) int* lptr_i32;

DEV_INLINE float neg_inf() { return -__builtin_inff(); }

DEV_INLINE float wave_max(float m) {
#pragma unroll
  for (int off = 16; off > 0; off >>= 1) m = fmaxf(m, __shfl_xor(m, off, 32));
  return m;
}
DEV_INLINE float wave_sum(float s) {
#pragma unroll
  for (int off = 16; off > 0; off >>= 1) s += __shfl_xor(s, off, 32);
  return s;
}

// Stable logaddexp with -inf handling (matches jnp.logaddexp semantics here)
DEV_INLINE float log_add_exp(float a, float b) {
  float m = fmaxf(a, b);
  if (m == neg_inf()) return m;
  return m + log1pf(expf(fminf(a, b) - m));
}

DEV_INLINE void stage_to_lds(float* dst, const float* src, int n, int tid, int nthr) {
#if defined(HAVE_ASYNC_B128)
  const int n4 = n >> 2;  // 16-byte chunks; dst/src bases are 16B aligned
  for (int i = tid; i < n4; i += nthr) {
    __builtin_amdgcn_global_load_async_to_lds_b128((gptr_v4i)(src + 4 * i),
                                                   (lptr_v4i)(dst + 4 * i), 0, 0);
  }
  for (int i = (n4 << 2) + tid; i < n; i += nthr) {
#  if defined(HAVE_ASYNC_B32)
    __builtin_amdgcn_global_load_async_to_lds_b32((gptr_i32)(src + i),
                                                  (lptr_i32)(dst + i), 0, 0);
#  else
    dst[i] = src[i];
#  endif
  }
#else
  for (int i = tid; i < n; i += nthr) dst[i] = src[i];
#endif
}

DEV_INLINE void wait_async_lds() {
#if defined(HAVE_WAIT_ASYNC)
  __builtin_amdgcn_s_wait_asynccnt(0);
#elif defined(HAVE_ASYNC_B128) || defined(HAVE_ASYNC_B32)
  asm volatile("s_wait_asynccnt 0" ::: "memory");
#endif
}

// ---------------------------------------------------------------------------
// Kernel 1: per-(b,u,{blank|target}) column LSE over T and normalization.
// Only 800 columns of h are ever needed (log_softmax is over axis=1 == T).
// One wave32 per column.
// ---------------------------------------------------------------------------
__global__ __launch_bounds__(32) void k_lse_norm(const float* __restrict__ h,
                                                 const int* __restrict__ targets,
                                                 float* __restrict__ blank_o,
                                                 float* __restrict__ emit_o) {
  const int w = blockIdx.x;          // 0 .. B*U*2-1
  const int b = w / (2 * U);
  const int r = w - b * 2 * U;
  const int u = r >> 1;
  const bool is_emit = (r & 1) != 0;
  if (is_emit && u == U - 1) return;  // emit only defined for u < U-1
  const int v = is_emit ? targets[b * (U - 1) + u] : 0;

  const size_t stride = (size_t)U * V;                       // t-stride in h
  const size_t base = (size_t)b * T * stride + (size_t)u * V + (size_t)v;
  const int lane = threadIdx.x;

  float vals[7];
  float m = neg_inf();
#pragma unroll
  for (int k = 0; k < 7; ++k) {
    int t = lane + 32 * k;
    float x = neg_inf();
    if (t < T) x = h[base + (size_t)t * stride];
    vals[k] = x;
    m = fmaxf(m, x);
  }
  m = wave_max(m);

  float s = 0.f;
#pragma unroll
  for (int k = 0; k < 7; ++k) {
    int t = lane + 32 * k;
    if (t < T) s += expf(vals[k] - m);
  }
  s = wave_sum(s);
  const float lse = m + logf(s);

  if (!is_emit) {
    float* out = blank_o + (size_t)b * NB + u;  // layout [b][t][u]
#pragma unroll
    for (int k = 0; k < 7; ++k) {
      int t = lane + 32 * k;
      if (t < T) out[(size_t)t * U] = vals[k] - lse;
    }
  } else {
    float* out = emit_o + (size_t)b * EM_STRIDE + u;  // layout [b][t][u], padded batch stride
#pragma unroll
    for (int k = 0; k < 7; ++k) {
      int t = lane + 32 * k;
      if (t < T) out[(size_t)t * (U - 1)] = vals[k] - lse;
    }
  }
}

// ---------------------------------------------------------------------------
// Kernel 2: RNN-T alpha DP, one block per batch, anti-diagonal wavefront.
// blank/emit staged into LDS (async global->LDS when available).
// alpha[t][u] = lae(alpha[t-1][u] + blank[t-1][u], alpha[t][u-1] + emit[t][u-1])
// ---------------------------------------------------------------------------
__global__ __launch_bounds__(64) void k_dp(const float* __restrict__ blank,
                                           const float* __restrict__ emit,
                                           const int* __restrict__ input_lens,
                                           const int* __restrict__ target_lens,
                                           float* __restrict__ loss) {
  __shared__ __align__(16) float s_bl[NB];   // 40.0 KB
  __shared__ __align__(16) float s_em[NE];   // 39.2 KB
  __shared__ float s_dg[2][64];              // diagonal ping-pong

  const int b = blockIdx.x;
  const int tid = threadIdx.x;
  const int nthr = blockDim.x;

  stage_to_lds(s_bl, blank + (size_t)b * NB, NB, tid, nthr);
  stage_to_lds(s_em, emit + (size_t)b * EM_STRIDE, NE, tid, nthr);
  wait_async_lds();
  __syncthreads();

  const int ti = input_lens[b] - 1;
  const int ui = target_lens[b] - 1;

  // diagonal d=0: alpha[0][0] = 0
  if (tid < U) s_dg[0][tid] = (tid == 0) ? 0.f : neg_inf();
  if (tid == 0 && ti == 0 && ui == 0) loss[b] = -(0.f + s_bl[0]);
  __syncthreads();

  for (int d = 1; d <= T + U - 2; ++d) {
    const int cu = d & 1, pu = cu ^ 1;
    if (tid < U) {
      const int u = tid;
      const int t = d - u;
      float val = neg_inf();
      if (t >= 0 && t < T) {
        float va = neg_inf(), vb = neg_inf();
        if (t >= 1) va = s_dg[pu][u] + s_bl[(t - 1) * U + u];           // from alpha[t-1][u]
        if (u >= 1) vb = s_dg[pu][u - 1] + s_em[t * (U - 1) + (u - 1)]; // from alpha[t][u-1]
        val = log_add_exp(va, vb);
        if (t == ti && u == ui) loss[b] = -(val + s_bl[ti * U + ui]);
      }
      s_dg[cu][u] = val;
    }
    __syncthreads();
  }
}

// ---------------------------------------------------------------------------
// Kernel 3: mean over batch -> d_out[0]
// ---------------------------------------------------------------------------
__global__ __launch_bounds__(32) void k_mean(const float* __restrict__ loss,
                                             float* __restrict__ out) {
  float v = (threadIdx.x < B) ? loss[threadIdx.x] : 0.f;
  v = wave_sum(v);
  if (threadIdx.x == 0) out[0] = v * (1.0f / B);
}

extern "C" void kernel_launch(void* const* d_in, const int* in_sizes, int n_in,
                              void* d_out, int out_size, void* d_ws, size_t ws_size,
                              hipStream_t stream) {
  const float* h          = (const float*)d_in[0];
  const int* targets      = (const int*)d_in[1];
  const int* input_lens   = (const int*)d_in[2];
  const int* target_lens  = (const int*)d_in[3];

  // Workspace layout (floats): blank[B][T][U] | emit[B][EM_STRIDE] | loss[B]
  float* blank = (float*)d_ws;
  float* emit  = blank + (size_t)B * NB;
  float* loss  = emit + (size_t)B * EM_STRIDE;

  k_lse_norm<<<B * U * 2, 32, 0, stream>>>(h, targets, blank, emit);
  k_dp<<<B, 64, 0, stream>>>(blank, emit, input_lens, target_lens, loss);
  k_mean<<<1, 32, 0, stream>>>(loss, (float*)d_out);

  (void)in_sizes; (void)n_in; (void)out_size; (void)ws_size;
}